// GraphConvolution_12945031430406
// MI455X (gfx1250) — compile-verified
//
#include <hip/hip_runtime.h>
#include <hip/hip_bf16.h>

typedef float v2f __attribute__((ext_vector_type(2)));
typedef float v8f __attribute__((ext_vector_type(8)));

#define IN_F 256
#define OUT_F 64

// ---------------------------------------------------------------------------
// Zero-fill the output (harness poisons d_out; segment_sum semantics need 0).
// ---------------------------------------------------------------------------
__global__ __launch_bounds__(256)
void gcn_zero_f32(float* __restrict__ p, int n) {
  int i = blockIdx.x * 256 + threadIdx.x;
  if (i < n) p[i] = 0.0f;
}

// ---------------------------------------------------------------------------
// h = x @ W + b using V_WMMA_F32_16X16X4_F32 (full fp32 precision).
// One wave32 per 16-row stripe, computing all 64 output columns (4 N-tiles)
// so the A fragment is reused 4x per K-step.
//
// Fragment layouts (CDNA5 ISA 7.12.2, 32-bit):
//   A 16x4 : lane l -> M = l%16 ; VGPR0/1 hold K = khalf, khalf+1
//            where khalf = 0 (lanes 0-15) or 2 (lanes 16-31)
//   B 4x16 : lane l -> N = l%16 ; VGPR0/1 hold K = khalf, khalf+1
//   C/D    : VGPR r, lane l -> M = r + (l>=16 ? 8 : 0), N = l%16
// ---------------------------------------------------------------------------
__global__ __launch_bounds__(256)
void gcn_gemm_wmma(const float* __restrict__ x, const float* __restrict__ W,
                   const float* __restrict__ bias, float* __restrict__ h,
                   int n_nodes) {
  const int lane  = threadIdx.x & 31;
  const int wave  = blockIdx.x * (blockDim.x >> 5) + (threadIdx.x >> 5);
  const int row0  = wave * 16;
  if (row0 + 16 > n_nodes) return;

  const int m     = lane & 15;          // row within tile (A), col within tile (B,C,D)
  const int khalf = (lane >> 4) << 1;   // 0 or 2 : K split across half-waves
  const int rhi   = (lane >> 4) << 3;   // +8 rows for upper half-wave in C/D

  const float* xrow = x + (size_t)(row0 + m) * IN_F + khalf;  // 8B-aligned
  const float* wcol = W + m;                                  // this lane's column

  v8f acc0 = {}, acc1 = {}, acc2 = {}, acc3 = {};

  for (int k0 = 0; k0 < IN_F; k0 += 4) {
    // A fragment: two consecutive K values for this lane -> single b64 load
    v2f a = *(const v2f*)(xrow + k0);

    // B fragments for the four 16-wide N-tiles
    const float* wk = wcol + (size_t)(k0 + khalf) * OUT_F;
    v2f b0, b1, b2, b3;
    b0.x = wk[0];          b0.y = wk[OUT_F + 0];
    b1.x = wk[16];         b1.y = wk[OUT_F + 16];
    b2.x = wk[32];         b2.y = wk[OUT_F + 32];
    b3.x = wk[48];         b3.y = wk[OUT_F + 48];

    // 8 args: (neg_a, A, neg_b, B, c_mod, C, reuse_a, reuse_b)
    acc0 = __builtin_amdgcn_wmma_f32_16x16x4_f32(false, a, false, b0, (short)0, acc0, false, false);
    acc1 = __builtin_amdgcn_wmma_f32_16x16x4_f32(false, a, false, b1, (short)0, acc1, false, false);
    acc2 = __builtin_amdgcn_wmma_f32_16x16x4_f32(false, a, false, b2, (short)0, acc2, false, false);
    acc3 = __builtin_amdgcn_wmma_f32_16x16x4_f32(false, a, false, b3, (short)0, acc3, false, false);
  }

  // Store D + bias. lane -> col (m), VGPR index r -> row offset.
  const float bv0 = bias[ 0 + m];
  const float bv1 = bias[16 + m];
  const float bv2 = bias[32 + m];
  const float bv3 = bias[48 + m];
#pragma unroll
  for (int r = 0; r < 8; ++r) {
    float* hp = h + (size_t)(row0 + r + rhi) * OUT_F + m;
    hp[ 0] = acc0[r] + bv0;
    hp[16] = acc1[r] + bv1;
    hp[32] = acc2[r] + bv2;
    hp[48] = acc3[r] + bv3;
  }
}

// ---------------------------------------------------------------------------
// out[dst] += w * h[src] : 16 threads per edge, float4 gather (L2-resident h),
// 4x global_atomic_add_f32 (L2-resident out, 12.8 MB << 192 MB L2).
// ---------------------------------------------------------------------------
__global__ __launch_bounds__(256)
void gcn_scatter(const float* __restrict__ h, const int* __restrict__ esrc,
                 const int* __restrict__ edst, const float* __restrict__ ew,
                 float* __restrict__ out, int n_edges) {
  const long long t = (long long)blockIdx.x * blockDim.x + threadIdx.x;
  const int e = (int)(t >> 4);
  if (e >= n_edges) return;
  const int f = ((int)t & 15) << 2;       // feature group of 4

  const int   s = esrc[e];
  const int   d = edst[e];
  const float w = ew[e];

  const float4 hv = *(const float4*)(h + (size_t)s * OUT_F + f);
  float* op = out + (size_t)d * OUT_F + f;
  atomicAdd(op + 0, w * hv.x);
  atomicAdd(op + 1, w * hv.y);
  atomicAdd(op + 2, w * hv.z);
  atomicAdd(op + 3, w * hv.w);
}

// ---------------------------------------------------------------------------
extern "C" void kernel_launch(void* const* d_in, const int* in_sizes, int n_in,
                              void* d_out, int out_size, void* d_ws, size_t ws_size,
                              hipStream_t stream) {
  const float* x    = (const float*)d_in[0];
  const int*   esrc = (const int*)  d_in[1];
  const int*   edst = (const int*)  d_in[2];
  const float* ew   = (const float*)d_in[3];
  const float* W    = (const float*)d_in[4];
  const float* b    = (const float*)d_in[5];
  float* out = (float*)d_out;
  float* h   = (float*)d_ws;   // [n_nodes, OUT_F] fp32 scratch (12.8 MB)

  const int n_nodes = in_sizes[0] / IN_F;
  const int n_edges = in_sizes[1];

  // 1) zero output accumulator
  gcn_zero_f32<<<(out_size + 255) / 256, 256, 0, stream>>>(out, out_size);

  // 2) h = x @ W + b   (WMMA f32, one wave per 16-row stripe, 8 waves/block)
  const int waves  = (n_nodes + 15) / 16;
  const int blocks = (waves + 7) / 8;
  gcn_gemm_wmma<<<blocks, 256, 0, stream>>>(x, W, b, h, n_nodes);

  // 3) out[dst] += w * h[src]
  const long long sthreads = (long long)n_edges * 16;
  const int sblocks = (int)((sthreads + 255) / 256);
  gcn_scatter<<<sblocks, 256, 0, stream>>>(h, esrc, edst, ew, out, n_edges);
}